// SoftDecisionTree_25108378812447
// MI455X (gfx1250) — compile-verified
//
#include <hip/hip_runtime.h>
#include <hip/hip_bf16.h>

// ---------------- problem constants (from reference) ----------------
constexpr int BATCH    = 8192;
constexpr int DIM      = 2048;
constexpr int NCLASS   = 128;   // OUTPUT_DIM
constexpr int N_INNER  = 255;   // 2^8 - 1
constexpr int N_LEAF   = 256;   // 2^8
constexpr int NPAD     = 256;   // padded inner-node count for the GEMM
constexpr float LMBDA  = 0.1f;

typedef float v2f __attribute__((ext_vector_type(2)));
typedef float v8f __attribute__((ext_vector_type(8)));

// ---------------- workspace layout (floats) ----------------
constexpr size_t WS_P     = 0;                                  // BATCH*256
constexpr size_t WS_Q     = WS_P + (size_t)BATCH * NPAD;        // 256*128
constexpr size_t WS_LOGQ  = WS_Q + (size_t)N_LEAF * NCLASS;     // 256*128
constexpr size_t WS_NUM   = WS_LOGQ + (size_t)N_LEAF * NCLASS;  // 128
constexpr size_t WS_DEN   = WS_NUM + 128;                       // 128
constexpr size_t WS_LOSS  = WS_DEN + 128;                       // 1

// ---------------- kernel 0: zero the small accumulators ----------------
__global__ void init_acc_kernel(float* acc) {
    int i = threadIdx.x;
    if (i < 384) acc[i] = 0.0f;   // pen_num[128] + pen_den[128] + loss(+pad)
}

// ---------------- kernel 1: Q = softmax(leaf_params), logQ ----------------
__global__ __launch_bounds__(128) void softmax_kernel(const float* __restrict__ leaf,
                                                      float* __restrict__ Q,
                                                      float* __restrict__ logQ) {
    __shared__ float red[128];
    const int l = blockIdx.x;
    const int c = threadIdx.x;
    float v = leaf[l * NCLASS + c];
    red[c] = v; __syncthreads();
    for (int o = 64; o > 0; o >>= 1) { if (c < o) red[c] = fmaxf(red[c], red[c + o]); __syncthreads(); }
    float mx = red[0]; __syncthreads();
    float e = __expf(v - mx);
    red[c] = e; __syncthreads();
    for (int o = 64; o > 0; o >>= 1) { if (c < o) red[c] += red[c + o]; __syncthreads(); }
    float s = red[0];
    Q[l * NCLASS + c]    = e / s;
    logQ[l * NCLASS + c] = (v - mx) - __logf(s);
}

// ---------------- kernel 2: fp32 WMMA GEMM + sigmoid ----------------
// One wave computes a 32x32 tile (2x2 register tiles of 16x16) of
// logits = x @ W.T, then p = sigmoid(beta*(logit+b)).
// A (16x4 f32): VGPR0 holds K+0 (lanes 0-15) / K+2 (lanes 16-31), VGPR1 K+1/K+3.
// B (4x16 f32): same K split per lane-half, N = lane&15.
// C/D (16x16 f32): VGPR r -> M = r + 8*(lane>>4), N = lane&15.
// Column 255 (padded node) accumulates garbage from a clamped W row and is
// overwritten with p=0 in the epilogue; WMMA B-lane N only affects D-column N.
__global__ __launch_bounds__(32) void gemm_sigmoid_kernel(const float* __restrict__ x,
                                                          const float* __restrict__ W,
                                                          const float* __restrict__ bvec,
                                                          const float* __restrict__ beta,
                                                          float* __restrict__ p) {
    const int m0   = blockIdx.x * 32;
    const int n0   = blockIdx.y * 32;
    const int lane = threadIdx.x;
    const int half = lane >> 4;
    const int l15  = lane & 15;

    const int nA = n0 + l15;        // first N sub-tile column for this lane
    const int nB = n0 + 16 + l15;   // second N sub-tile column

    const float* __restrict__ arow0 = x + (size_t)(m0 + l15) * DIM;
    const float* __restrict__ arow1 = arow0 + (size_t)16 * DIM;
    const float* __restrict__ brow0 = W + (size_t)(nA < N_INNER ? nA : 0) * DIM;
    const float* __restrict__ brow1 = W + (size_t)(nB < N_INNER ? nB : 0) * DIM;

    v8f acc00 = {}, acc01 = {}, acc10 = {}, acc11 = {};
    #pragma unroll 2
    for (int k = 0; k < DIM; k += 8) {
        const int kk = k + 2 * half;
        v2f a00 = *(const v2f*)(arow0 + kk);
        v2f a01 = *(const v2f*)(arow0 + kk + 4);
        v2f a10 = *(const v2f*)(arow1 + kk);
        v2f a11 = *(const v2f*)(arow1 + kk + 4);
        v2f b00 = *(const v2f*)(brow0 + kk);
        v2f b01 = *(const v2f*)(brow0 + kk + 4);
        v2f b10 = *(const v2f*)(brow1 + kk);
        v2f b11 = *(const v2f*)(brow1 + kk + 4);
        acc00 = __builtin_amdgcn_wmma_f32_16x16x4_f32(false, a00, false, b00, (short)0, acc00, false, false);
        acc01 = __builtin_amdgcn_wmma_f32_16x16x4_f32(false, a00, false, b10, (short)0, acc01, false, false);
        acc10 = __builtin_amdgcn_wmma_f32_16x16x4_f32(false, a10, false, b00, (short)0, acc10, false, false);
        acc11 = __builtin_amdgcn_wmma_f32_16x16x4_f32(false, a10, false, b10, (short)0, acc11, false, false);
        acc00 = __builtin_amdgcn_wmma_f32_16x16x4_f32(false, a01, false, b01, (short)0, acc00, false, false);
        acc01 = __builtin_amdgcn_wmma_f32_16x16x4_f32(false, a01, false, b11, (short)0, acc01, false, false);
        acc10 = __builtin_amdgcn_wmma_f32_16x16x4_f32(false, a11, false, b01, (short)0, acc10, false, false);
        acc11 = __builtin_amdgcn_wmma_f32_16x16x4_f32(false, a11, false, b11, (short)0, acc11, false, false);
    }

    // epilogue: p = sigmoid(beta*(logit + b)); fast rcp is fine for a sigmoid
    #pragma unroll
    for (int ni = 0; ni < 2; ++ni) {
        const int   nn     = ni ? nB : nA;
        const bool  nvalid = (nn < N_INNER);
        const float bias   = nvalid ? bvec[nn] : 0.0f;
        const float bet    = nvalid ? beta[nn] : 0.0f;
        #pragma unroll
        for (int mi = 0; mi < 2; ++mi) {
            const v8f acc = ni ? (mi ? acc11 : acc01) : (mi ? acc10 : acc00);
            #pragma unroll
            for (int r = 0; r < 8; ++r) {
                const int   m  = m0 + 16 * mi + r + 8 * half;
                const float z  = bet * (acc[r] + bias);
                const float pr = nvalid ? __builtin_amdgcn_rcpf(1.0f + __expf(-z)) : 0.0f;
                p[(size_t)m * NPAD + nn] = pr;
            }
        }
    }
}

// ---------------- kernel 3: tree walk ----------------
// 256 threads = 8 waves; each wave handles one batch row per round, 4 rounds (32 rows/block).
// Lane owns leaves [8*lane, 8*lane+8): depth>=5 subtree sums are lane-local,
// depths 4..0 reduced with __shfl_xor across the wave32.
__global__ __launch_bounds__(256) void tree_kernel(const float* __restrict__ p,
                                                   const int*   __restrict__ target,
                                                   const float* __restrict__ Q,
                                                   const float* __restrict__ logQ,
                                                   float* __restrict__ pen_num,
                                                   float* __restrict__ pen_den,
                                                   float* __restrict__ loss_g,
                                                   float* __restrict__ out) {
    __shared__ float p_s[8][NPAD];
    __shared__ float num_s[128];
    __shared__ float den_s[128];
    __shared__ float loss_s;

    const int tid  = threadIdx.x;
    const int wv   = tid >> 5;
    const int lane = tid & 31;

    if (tid < 128) { num_s[tid] = 0.0f; den_s[tid] = 0.0f; }
    if (tid == 0)  loss_s = 0.0f;
    __syncthreads();

    float wloss = 0.0f;
    const int row0 = blockIdx.x * 32;

    for (int rr = 0; rr < 4; ++rr) {
        const int row = row0 + rr * 8 + wv;

        // stage this row's p values (same-wave LDS ops are in-order)
        const float* __restrict__ prow = p + (size_t)row * NPAD;
        #pragma unroll
        for (int j = 0; j < 8; ++j) p_s[wv][lane + 32 * j] = prow[lane + 32 * j];
        const int t = target[row];

        // shared prefix product over depths 0..4 (ancestors of this lane's 8 leaves)
        float prefix = 1.0f;
        #pragma unroll
        for (int d = 0; d < 5; ++d) {
            const int   node = (1 << d) - 1 + (lane >> (5 - d));
            const int   take = (lane >> (4 - d)) & 1;
            const float pv   = p_s[wv][node];
            prefix *= take ? pv : (1.0f - pv);
        }

        const float p5  = p_s[wv][31 + lane];
        const float p6a = p_s[wv][63 + 2 * lane];
        const float p6b = p_s[wv][63 + 2 * lane + 1];
        float p7[4];
        #pragma unroll
        for (int j = 0; j < 4; ++j) p7[j] = p_s[wv][127 + 4 * lane + j];

        float w[8];
        #pragma unroll
        for (int j = 0; j < 8; ++j) {
            const float f5  = (j & 4) ? p5 : (1.0f - p5);
            const float p6  = (j & 4) ? p6b : p6a;
            const float f6  = (j & 2) ? p6 : (1.0f - p6);
            const float pp7 = p7[j >> 1];
            const float f7  = (j & 1) ? pp7 : (1.0f - pp7);
            w[j] = prefix * f5 * f6 * f7;
        }

        // penalties: subtree sums (den) and p*den (num), depths 6..0
        const float s6a = w[0] + w[1] + w[2] + w[3];
        const float s6b = w[4] + w[5] + w[6] + w[7];
        atomicAdd(&den_s[63 + 2 * lane],     s6a);
        atomicAdd(&num_s[63 + 2 * lane],     p6a * s6a);
        atomicAdd(&den_s[63 + 2 * lane + 1], s6b);
        atomicAdd(&num_s[63 + 2 * lane + 1], p6b * s6b);

        float s = s6a + s6b;                 // depth-5 subtree sum, node 31+lane
        atomicAdd(&den_s[31 + lane], s);
        atomicAdd(&num_s[31 + lane], p5 * s);

        #pragma unroll
        for (int d = 4; d >= 0; --d) {
            s += __shfl_xor(s, 1 << (4 - d), 32);
            if ((lane & ((1 << (5 - d)) - 1)) == 0) {
                const int node = (1 << d) - 1 + (lane >> (5 - d));
                atomicAdd(&den_s[node], s);
                atomicAdd(&num_s[node], p_s[wv][node] * s);
            }
        }

        // loss contribution + first-max argmax
        float ls = 0.0f;
        float bv = -1.0f; int bi = 0;
        #pragma unroll
        for (int j = 0; j < 8; ++j) {
            const int l = lane * 8 + j;
            ls += w[j] * logQ[(size_t)l * NCLASS + t];
            if (w[j] > bv) { bv = w[j]; bi = l; }
        }
        #pragma unroll
        for (int off = 16; off >= 1; off >>= 1) ls += __shfl_xor(ls, off, 32);
        #pragma unroll
        for (int off = 16; off >= 1; off >>= 1) {
            const float ov = __shfl_xor(bv, off, 32);
            const int   oi = __shfl_xor(bi, off, 32);
            if (ov > bv || (ov == bv && oi < bi)) { bv = ov; bi = oi; }
        }
        wloss += ls;                          // only lane 0's copy is used

        // output row = Q[argmax leaf]
        float* __restrict__ orow = out + (size_t)row * NCLASS;
        const float* __restrict__ qrow = Q + (size_t)bi * NCLASS;
        for (int c = lane; c < NCLASS; c += 32) orow[c] = qrow[c];
    }

    if (lane == 0) atomicAdd(&loss_s, wloss);
    __syncthreads();
    if (tid < 127) {
        atomicAdd(&pen_num[tid], num_s[tid]);
        atomicAdd(&pen_den[tid], den_s[tid]);
    }
    if (tid == 255) atomicAdd(loss_g, loss_s);
}

// ---------------- kernel 4: finalize scalar ----------------
__global__ __launch_bounds__(128) void finalize_kernel(const float* __restrict__ pen_num,
                                                       const float* __restrict__ pen_den,
                                                       const float* __restrict__ loss_g,
                                                       float* __restrict__ out_total) {
    __shared__ float red[128];
    const int t = threadIdx.x;
    float c = 0.0f;
    if (t < 127) {                      // nodes 0..126 (depths 0..6)
        const int   d   = 31 - __clz(t + 1);
        const float lam = LMBDA * exp2f(-(float)(d + 1));
        const float pen = pen_num[t] / pen_den[t];
        c = -lam * 0.5f * (__logf(pen) + __logf(1.0f - pen));
    }
    red[t] = c; __syncthreads();
    for (int o = 64; o > 0; o >>= 1) { if (t < o) red[t] += red[t + o]; __syncthreads(); }
    if (t == 0) out_total[0] = -(loss_g[0] / (float)BATCH) + red[0];
}

// ---------------- launcher ----------------
extern "C" void kernel_launch(void* const* d_in, const int* in_sizes, int n_in,
                              void* d_out, int out_size, void* d_ws, size_t ws_size,
                              hipStream_t stream) {
    const float* x      = (const float*)d_in[0];
    const int*   target = (const int*)  d_in[1];
    const float* W      = (const float*)d_in[2];
    const float* bvec   = (const float*)d_in[3];
    const float* beta   = (const float*)d_in[4];
    const float* leaf   = (const float*)d_in[5];

    float* ws      = (float*)d_ws;
    float* p       = ws + WS_P;
    float* Q       = ws + WS_Q;
    float* logQ    = ws + WS_LOGQ;
    float* pen_num = ws + WS_NUM;
    float* pen_den = ws + WS_DEN;
    float* loss_g  = ws + WS_LOSS;

    float* out   = (float*)d_out;        // [0] = total, [1..] = output (8192x128)

    init_acc_kernel<<<1, 384, 0, stream>>>(pen_num);
    softmax_kernel<<<N_LEAF, 128, 0, stream>>>(leaf, Q, logQ);
    gemm_sigmoid_kernel<<<dim3(BATCH / 32, NPAD / 32), 32, 0, stream>>>(x, W, bvec, beta, p);
    tree_kernel<<<BATCH / 32, 256, 0, stream>>>(p, target, Q, logQ, pen_num, pen_den, loss_g, out + 1);
    finalize_kernel<<<1, 128, 0, stream>>>(pen_num, pen_den, loss_g, out);
}